// DualAxisMemoryAttention_89455578841863
// MI455X (gfx1250) — compile-verified
//
#include <hip/hip_runtime.h>
#include <math.h>

typedef __attribute__((ext_vector_type(2))) float v2f;
typedef __attribute__((ext_vector_type(8))) float v8f;
typedef __attribute__((ext_vector_type(4))) unsigned int u32x4;
typedef __attribute__((ext_vector_type(8))) int i32x8;
typedef __attribute__((ext_vector_type(4))) int i32x4;

#define D_MODEL 1024
#define NHEADS  16
#define HDIM    64
#define SEQ     2048
#define BATCH   4
#define NLAYERS 12
#define MTOT    (BATCH*SEQ)      /* 8192 rows */
#define SCALE   0.125f           /* 1/sqrt(64) */
#define NEG_BIG (-1e30f)

static __device__ __forceinline__ v8f wmma4(v2f a, v2f b, v8f c) {
  // D = A(16x4,f32) * B(4x16,f32) + C(16x16,f32)
  return __builtin_amdgcn_wmma_f32_16x16x4_f32(false, a, false, b, (short)0, c,
                                               false, false);
}

// ---------------------------------------------------------------------------
// Generic projection GEMM:  C[M,N] = A[M,K=1024] @ W[N,K]^T + bias
// One wave computes a 32(M) x 64(N) tile: 2 A-frags share each B-frag,
// 8 f32 WMMA accumulators (8 WMMA per 6 b64 loads).
// MODE 0: row-major store to dst0
// MODE 1: qkv scatter: n<1024 -> dst0 (q_row), n<2048 -> dst1 (k), else dst2 (v)
//         all in (B,H,S,Hd) layout
// MODE 2: head-scatter single tensor (q_col) in (B,H,S,Hd) layout
// ---------------------------------------------------------------------------
template<int MODE>
__global__ __launch_bounds__(128)
void gemm_kernel(const float* __restrict__ A, const float* __restrict__ W,
                 const float* __restrict__ bias,
                 float* __restrict__ dst0, float* __restrict__ dst1,
                 float* __restrict__ dst2, int N)
{
  const int K = D_MODEL;
  const int wave = threadIdx.x >> 5;
  const int lane = threadIdx.x & 31;
  const int lo = lane & 15, hi = lane >> 4;
  const int tile = blockIdx.x * 4 + wave;
  const int ntn = N >> 6;
  const int mt = tile / ntn, nt = tile - mt * ntn;
  const int m0 = mt << 5, n0 = nt << 6;

  v8f acc[2][4] = {};
  const float* Arow0 = A + (size_t)(m0 + lo) * K;
  const float* Arow1 = A + (size_t)(m0 + 16 + lo) * K;
  for (int k0 = 0; k0 < K; k0 += 4) {
    v2f a0 = *(const v2f*)(Arow0 + k0 + 2 * hi);
    v2f a1 = *(const v2f*)(Arow1 + k0 + 2 * hi);
#pragma unroll
    for (int f = 0; f < 4; ++f) {
      const float* Wp = W + (size_t)(n0 + f * 16 + lo) * K + k0 + 2 * hi;
      v2f b = *(const v2f*)Wp;
      acc[0][f] = wmma4(a0, b, acc[0][f]);
      acc[1][f] = wmma4(a1, b, acc[1][f]);
    }
  }

#pragma unroll
  for (int mi = 0; mi < 2; ++mi) {
#pragma unroll
    for (int f = 0; f < 4; ++f) {
      int n = n0 + f * 16 + lo;
      float bv = bias[n];
#pragma unroll
      for (int r = 0; r < 8; ++r) {
        float val = acc[mi][f][r] + bv;
        int m = m0 + mi * 16 + r + 8 * hi;   // C layout: VGPR r -> rows r/r+8
        if (MODE == 0) {
          dst0[(size_t)m * N + n] = val;
        } else {
          int which = n >> 10;
          int nl = n & 1023;
          int h = nl >> 6, hd = nl & 63;
          int bb = m >> 11, s = m & 2047;
          float* dp = (MODE == 2) ? dst0
                    : (which == 0 ? dst0 : (which == 1 ? dst1 : dst2));
          dp[((size_t)(bb * NHEADS + h) * SEQ + s) * HDIM + hd] = val;
        }
      }
    }
  }
}

// ---------------------------------------------------------------------------
// LayerNorm statistics over D for each (l,b,s) row of past_states.
// One wave per 1024-float row; 98304 rows total.
// ---------------------------------------------------------------------------
__global__ __launch_bounds__(256)
void ln_stats_kernel(const float* __restrict__ past,
                     float* __restrict__ mu, float* __restrict__ rstd)
{
  int wave = threadIdx.x >> 5, lane = threadIdx.x & 31;
  int row = blockIdx.x * 8 + wave;
  const float4* p = (const float4*)(past + (size_t)row * D_MODEL);
  float s = 0.f, ss = 0.f;
#pragma unroll
  for (int j = 0; j < 8; ++j) {
    float4 x = p[lane + j * 32];
    s  += x.x + x.y + x.z + x.w;
    ss += x.x * x.x + x.y * x.y + x.z * x.z + x.w * x.w;
  }
  for (int msk = 1; msk < 32; msk <<= 1) {
    s  += __shfl_xor(s,  msk, 32);
    ss += __shfl_xor(ss, msk, 32);
  }
  if (lane == 0) {
    float m = s * (1.0f / 1024.0f);
    float v = ss * (1.0f / 1024.0f) - m * m;
    mu[row]   = m;
    rstd[row] = rsqrtf(v + 1e-5f);
  }
}

// ---------------------------------------------------------------------------
// Fused flash attention + per-position memory attention.
// One wave per (b, h, 16-row q tile). Causal over S in 32-key steps, plus
// 12 memory slots folded into the same online softmax.
// Memory-bank slice (12x16x64 f32, 48KB) is DMA'd into LDS by the Tensor
// Data Mover as a 3-D tile, then normalized in place.
// ---------------------------------------------------------------------------
__global__ __launch_bounds__(32)
void attn_kernel(const float* __restrict__ qrow, const float* __restrict__ kk,
                 const float* __restrict__ vv,  const float* __restrict__ qcol,
                 const float* __restrict__ past, const float* __restrict__ mu,
                 const float* __restrict__ rstd, float* __restrict__ merged)
{
  __shared__ float mbn[NLAYERS * 16 * HDIM];  // normalized memory bank, 48 KB
  __shared__ float pt[16 * 32];               // P bounce / score routing
  __shared__ float muS[NLAYERS * 16];
  __shared__ float rsS[NLAYERS * 16];

  const int lane = threadIdx.x, lo = lane & 15, hi = lane >> 4;
  const int qt = blockIdx.x & 127;
  const int bh = blockIdx.x >> 7;
  const int h = bh & 15, b = bh >> 4;
  const int s0 = qt * 16;
  const size_t bhS = (size_t)bh * SEQ;

  // ---- TDM: async 3-D tile load of past_states slice into LDS ----
  // tile: dim0 = 64 floats (contiguous), dim1 = 16 rows (stride 1024),
  //       dim2 = 12 layers (stride B*S*D = 8388608)
  {
    unsigned int lds_off = (unsigned int)(uintptr_t)(void*)mbn; // addr[31:0]
    unsigned long long ga = (unsigned long long)(uintptr_t)past +
        ((((unsigned long long)b * SEQ + s0) * D_MODEL) + h * HDIM) * 4ull;
    u32x4 g0;
    g0[0] = 1u;                                   // count=1, user descriptor
    g0[1] = lds_off;                              // lds_addr
    g0[2] = (unsigned int)(ga & 0xFFFFFFFFu);     // global_addr[31:0]
    g0[3] = (unsigned int)((ga >> 32) & 0x01FFFFFFu) | (2u << 30); // type=2
    i32x8 g1;
    g1[0] = 0x20000;                // data_size = 2 (4 bytes)
    g1[1] = (int)(64u << 16);       // tensor_dim0 = 64
    g1[2] = (int)(2048u << 16);     // tensor_dim1 = 2048
    g1[3] = (int)(64u << 16);       // tile_dim0 = 64
    g1[4] = (int)(16u | (12u << 16)); // tile_dim1 = 16, tile_dim2 = 12
    g1[5] = 1024;                   // tensor_dim0_stride = 1024
    g1[6] = 0;                      // dim0_stride hi / dim1_stride lo
    g1[7] = 0x80;                   // tensor_dim1_stride = 0x800000 (>>16)
    i32x4 g2; g2[0] = 12; g2[1] = 0; g2[2] = 0; g2[3] = 0; // tensor_dim2 = 12
    i32x4 g3; g3[0] = 0;  g3[1] = 0; g3[2] = 0; g3[3] = 0;
    i32x8 g4; g4[0] = 0; g4[1] = 0; g4[2] = 0; g4[3] = 0;
    g4[4] = 0; g4[5] = 0; g4[6] = 0; g4[7] = 0;
    __builtin_amdgcn_tensor_load_to_lds(g0, g1, g2, g3, g4, 0);
  }

  // ---- overlap with DMA: stage LN stats into LDS ----
  for (int i = lane; i < NLAYERS * 16; i += 32) {
    int l = i >> 4, row = i & 15;
    int sidx = (l * BATCH + b) * SEQ + s0 + row;
    muS[i] = mu[sidx];
    rsS[i] = rstd[sidx];
  }

  // ---- overlap with DMA: load Q tile as 16 A-fragments ----
  v2f qa[16];
  {
    size_t qb = (bhS + s0 + lo) * HDIM;
#pragma unroll
    for (int ksi = 0; ksi < 16; ++ksi)
      qa[ksi] = *(const v2f*)(qrow + qb + ksi * 4 + 2 * hi);
  }
  // q_col for row=lo, hd half owned by this lane
  float4 qc[8];
  {
    size_t cb = (bhS + s0 + lo) * HDIM + hi * 32;
#pragma unroll
    for (int i = 0; i < 8; ++i) qc[i] = *(const float4*)(qcol + cb + i * 4);
  }

  // ---- wait for tensor DMA, then normalize bank in place ----
  __builtin_amdgcn_s_wait_tensorcnt(0);
  __syncthreads();
  for (int i4 = lane; i4 < (NLAYERS * 16 * HDIM) / 4; i4 += 32) {
    int e = i4 * 4;
    int l = e >> 10, row = (e >> 6) & 15;
    float m = muS[l * 16 + row], r = rsS[l * 16 + row];
    float4 x = *(float4*)(mbn + e);
    x.x = (x.x - m) * r; x.y = (x.y - m) * r;
    x.z = (x.z - m) * r; x.w = (x.w - m) * r;
    *(float4*)(mbn + e) = x;
  }
  __syncthreads();

  v8f o[4] = {};
  float mr[8], lr[8];
#pragma unroll
  for (int r = 0; r < 8; ++r) { mr[r] = NEG_BIG; lr[r] = 0.f; }

  // ---- causal flash loop, 32 keys per iteration ----
  const int npair = (qt + 2) >> 1;   // ceil((qt+1)/2)
  for (int j2 = 0; j2 < npair; ++j2) {
    const int t0 = 2 * j2, t1 = 2 * j2 + 1;
    const bool has1 = (t1 <= qt);
    v8f sc0 = {}, sc1 = {};
    {
      size_t kb = (bhS + t0 * 16 + lo) * HDIM + 2 * hi;
#pragma unroll
      for (int ksi = 0; ksi < 16; ++ksi)
        sc0 = wmma4(qa[ksi], *(const v2f*)(kk + kb + ksi * 4), sc0);
    }
    if (has1) {
      size_t kb = (bhS + t1 * 16 + lo) * HDIM + 2 * hi;
#pragma unroll
      for (int ksi = 0; ksi < 16; ++ksi)
        sc1 = wmma4(qa[ksi], *(const v2f*)(kk + kb + ksi * 4), sc1);
    }
    float va0[8], va1[8];
#pragma unroll
    for (int r = 0; r < 8; ++r) {
      int rowi = r + 8 * hi;
      float x0 = sc0[r] * SCALE;
      if (t0 == qt && lo > rowi) x0 = NEG_BIG;
      va0[r] = x0;
      float x1 = has1 ? sc1[r] * SCALE : NEG_BIG;
      if (t1 == qt && lo > rowi) x1 = NEG_BIG;
      va1[r] = x1;
    }
#pragma unroll
    for (int r = 0; r < 8; ++r) {
      float t = fmaxf(va0[r], va1[r]);
      t = fmaxf(t, __shfl_xor(t, 1, 32));
      t = fmaxf(t, __shfl_xor(t, 2, 32));
      t = fmaxf(t, __shfl_xor(t, 4, 32));
      t = fmaxf(t, __shfl_xor(t, 8, 32));
      float mnew  = fmaxf(mr[r], t);
      float alpha = __expf(mr[r] - mnew);
      mr[r] = mnew;
      float p0 = __expf(va0[r] - mnew);
      float p1 = __expf(va1[r] - mnew);
      float su = p0 + p1;
      su += __shfl_xor(su, 1, 32); su += __shfl_xor(su, 2, 32);
      su += __shfl_xor(su, 4, 32); su += __shfl_xor(su, 8, 32);
      lr[r] = lr[r] * alpha + su;
      o[0][r] *= alpha; o[1][r] *= alpha; o[2][r] *= alpha; o[3][r] *= alpha;
      int rowi = r + 8 * hi;
      pt[rowi * 32 + lo]      = p0;   // bounce P: C layout -> A layout
      pt[rowi * 32 + 16 + lo] = p1;
    }
    __syncthreads();
#pragma unroll
    for (int ks2 = 0; ks2 < 4; ++ks2) {
      int kx = ks2 * 4 + 2 * hi;
      v2f af; af.x = pt[lo * 32 + kx]; af.y = pt[lo * 32 + kx + 1];
      size_t v0 = (bhS + t0 * 16 + kx) * HDIM;
#pragma unroll
      for (int f = 0; f < 4; ++f) {
        v2f bf; bf.x = vv[v0 + f * 16 + lo]; bf.y = vv[v0 + HDIM + f * 16 + lo];
        o[f] = wmma4(af, bf, o[f]);
      }
    }
    if (has1) {
#pragma unroll
      for (int ks2 = 0; ks2 < 4; ++ks2) {
        int kx = ks2 * 4 + 2 * hi;
        v2f af; af.x = pt[lo * 32 + 16 + kx]; af.y = pt[lo * 32 + 16 + kx + 1];
        size_t v0 = (bhS + t1 * 16 + kx) * HDIM;
#pragma unroll
        for (int f = 0; f < 4; ++f) {
          v2f bf; bf.x = vv[v0 + f * 16 + lo]; bf.y = vv[v0 + HDIM + f * 16 + lo];
          o[f] = wmma4(af, bf, o[f]);
        }
      }
    }
    __syncthreads();
  }

  // ---- memory-axis scores: 12 slots for row = lo (split over hd halves) ----
  float msc[NLAYERS];
#pragma unroll
  for (int l = 0; l < NLAYERS; ++l) {
    const float* mp = mbn + (l * 16 + lo) * HDIM + hi * 32;
    float part = 0.f;
#pragma unroll
    for (int i = 0; i < 8; ++i) {
      float4 m4 = *(const float4*)(mp + i * 4);
      float4 q4 = qc[i];
      part += m4.x * q4.x + m4.y * q4.y + m4.z * q4.z + m4.w * q4.w;
    }
    part += __shfl_xor(part, 16, 32);
    msc[l] = part * SCALE;
  }
  __syncthreads();
#pragma unroll
  for (int l = 0; l < NLAYERS; ++l) pt[lo * NLAYERS + l] = msc[l]; // dup-safe
  __syncthreads();

  // ---- fold memory slots into softmax state + accumulate memory context ----
#pragma unroll
  for (int r = 0; r < 8; ++r) {
    int rowi = r + 8 * hi;
    float sc12[NLAYERS], t = NEG_BIG;
#pragma unroll
    for (int l = 0; l < NLAYERS; ++l) {
      sc12[l] = pt[rowi * NLAYERS + l];
      t = fmaxf(t, sc12[l]);
    }
    float mnew  = fmaxf(mr[r], t);
    float alpha = __expf(mr[r] - mnew);
    float su = 0.f, w12[NLAYERS];
#pragma unroll
    for (int l = 0; l < NLAYERS; ++l) { w12[l] = __expf(sc12[l] - mnew); su += w12[l]; }
    lr[r] = lr[r] * alpha + su;
#pragma unroll
    for (int f = 0; f < 4; ++f) {
      float a0 = o[f][r] * alpha;
#pragma unroll
      for (int l = 0; l < NLAYERS; ++l)
        a0 += w12[l] * mbn[(l * 16 + rowi) * HDIM + f * 16 + lo];
      o[f][r] = a0;
    }
  }

  // ---- normalize and store merged (B,S,D) ----
#pragma unroll
  for (int r = 0; r < 8; ++r) {
    int rowi = r + 8 * hi;
    float invl = 1.0f / lr[r];
    size_t ob = ((size_t)(b * SEQ + s0 + rowi)) * D_MODEL + h * HDIM;
#pragma unroll
    for (int f = 0; f < 4; ++f)
      merged[ob + f * 16 + lo] = o[f][r] * invl;
  }
}

// ---------------------------------------------------------------------------
extern "C" void kernel_launch(void* const* d_in, const int* in_sizes, int n_in,
                              void* d_out, int out_size, void* d_ws, size_t ws_size,
                              hipStream_t stream)
{
  (void)in_sizes; (void)n_in; (void)out_size; (void)ws_size;
  const float* x      = (const float*)d_in[0];
  const float* past   = (const float*)d_in[1];
  const float* qkv_w  = (const float*)d_in[2];
  const float* qkv_b  = (const float*)d_in[3];
  const float* colq_w = (const float*)d_in[4];
  const float* colq_b = (const float*)d_in[5];
  const float* out_w  = (const float*)d_in[6];
  const float* out_b  = (const float*)d_in[7];

  const size_t TEN = (size_t)MTOT * D_MODEL;   // 8,388,608 floats
  float* out   = (float*)d_out;                // (B,S,D)
  float* kout  = out  + TEN;                   // (B,H,S,Hd)
  float* vout  = kout + TEN;
  float* qrout = vout + TEN;

  float* ws     = (float*)d_ws;
  float* qcol   = ws;                          // (B,H,S,Hd)
  float* merged = qcol + TEN;                  // (B,S,D)
  float* mu     = merged + TEN;                // L*B*S
  float* rstd   = mu + (size_t)NLAYERS * MTOT;

  // 1) qkv projection -> q_row/k/v directly in output regions (head layout)
  {
    int tiles = (MTOT / 32) * (3 * D_MODEL / 64);   // 256*48 = 12288
    gemm_kernel<1><<<tiles / 4, 128, 0, stream>>>(x, qkv_w, qkv_b,
                                                  qrout, kout, vout, 3 * D_MODEL);
  }
  // 2) column-query projection -> ws (head layout)
  {
    int tiles = (MTOT / 32) * (D_MODEL / 64);       // 4096
    gemm_kernel<2><<<tiles / 4, 128, 0, stream>>>(x, colq_w, colq_b,
                                                  qcol, nullptr, nullptr, D_MODEL);
  }
  // 3) LayerNorm stats of past_states
  ln_stats_kernel<<<(NLAYERS * MTOT) / 8, 256, 0, stream>>>(past, mu, rstd);
  // 4) fused flash + memory attention -> merged
  attn_kernel<<<BATCH * NHEADS * (SEQ / 16), 32, 0, stream>>>(
      qrout, kout, vout, qcol, past, mu, rstd, merged);
  // 5) output projection
  {
    int tiles = (MTOT / 32) * (D_MODEL / 64);       // 4096
    gemm_kernel<0><<<tiles / 4, 128, 0, stream>>>(merged, out_w, out_b,
                                                  out, nullptr, nullptr, D_MODEL);
  }
}